// CubeSimulatorSlabbed_90271622627819
// MI455X (gfx1250) — compile-verified
//
#include <hip/hip_runtime.h>
#include <math.h>

// ---------------------------------------------------------------------------
// CubeSimulatorSlabbed for MI455X (gfx1250, wave32)
//   Stage 1: transcendental-bound voxel sweep, one v_exp_f32 per (voxel,chan)
//   Stage 2/3: separable 2x2 spatial box filter as real GEMMs via
//              V_WMMA_F32_16X16X4_F32 (Out = P^T * M * P, P = 0.5-avg matrix)
// ---------------------------------------------------------------------------

#define NHI   128   // hi-res spatial grid (64 * 2)
#define NOUT  64
#define NFREQ 32
#define NFINE 128   // 32 * 4 fine freq channels

typedef __attribute__((ext_vector_type(2))) float v2f;
typedef __attribute__((ext_vector_type(8))) float v8f;

__device__ __forceinline__ float fast_exp2(float x) {
#if __has_builtin(__builtin_amdgcn_exp2f)
  return __builtin_amdgcn_exp2f(x);      // v_exp_f32
#else
  return exp2f(x);
#endif
}

// ---------------------------------------------------------------------------
// Stage 1: cube_mid[F][x][y] = (norm/4) * sum_{fine in F} sum_z inten * g
// Thread map: tid -> (x, c, y); c = which block of 32 fine channels (4 per px)
// ---------------------------------------------------------------------------
__global__ __launch_bounds__(256)
void cube_stage1(const float* __restrict__ pInc, const float* __restrict__ pRot,
                 const float* __restrict__ pSig, const float* __restrict__ pVsh,
                 const float* __restrict__ pVmax, const float* __restrict__ pRt,
                 const float* __restrict__ pRd,  const float* __restrict__ pHz,
                 const float* __restrict__ freqs, float* __restrict__ mid)
{
  const float C_KMS   = 299792.458f;
  const float F0      = 230538000000.0f;
  const float LN2_INV = 1.4426950408889634f;   // 1/ln(2)
  const float TWO_PI  = 6.283185307179586f;
  const float TWO_OVER_PI = 0.6366197723675814f;

  int tid = blockIdx.x * blockDim.x + threadIdx.x;
  int y = tid & (NHI - 1);
  int c = (tid >> 7) & 3;        // fine-channel chunk: [32c, 32c+32)
  int x = tid >> 9;              // 0..127

  float inc  = pInc[0],  rot   = pRot[0], sig = pSig[0], vsh = pVsh[0];
  float vmax = pVmax[0], rturn = pRt[0],  rdisk = pRd[0], hz = pHz[0];

  float ci = cosf(inc), si = sinf(inc);
  float cp = cosf(rot), sp = sinf(rot);

  const float dx = 1000.0f / 127.0f;           // 2*FOV_HALF/(NHI-1)
  float X = dx * ((float)x - 63.5f);
  float Y = dx * ((float)y - 63.5f);

  // fine-frequency -> velocity labels for this thread's 32 channels
  float df  = freqs[1] - freqs[0];
  float dff = 0.25f * df;
  float f0f = freqs[0] - 1.5f * dff;
  float vel[32];
#pragma unroll
  for (int j = 0; j < 32; ++j) {
    float ff = f0f + dff * (float)(c * 32 + j);
    vel[j] = C_KMS * (1.0f - ff / F0) - vsh;
  }

  float sig_sq = sig * sig;
  float kexp   = -0.5f * LN2_INV / sig_sq;     // exponent coeff in log2
  float irdisk = 1.0f / rdisk;
  float ihz    = 1.0f / hz;
  float irturn = 1.0f / rturn;
  float vfac   = vmax * TWO_OVER_PI;

  // z-invariant pieces of the rotation
  float rx  = cp * X - sp * Y;                 // R02 == 0
  float sxy = sp * X + cp * Y;

  float acc[32];
#pragma unroll
  for (int j = 0; j < 32; ++j) acc[j] = 0.0f;

  for (int z = 0; z < NHI; ++z) {
    float Z  = dx * ((float)z - 63.5f);
    float ry = ci * sxy - si * Z;
    float rz = si * sxy + ci * Z;
    float rr = rx * rx + ry * ry;
    float r  = sqrtf(rr);
    float rinv = (rr > 0.0f) ? (1.0f / r) : 0.0f;
    float vabs = vfac * atanf(r * irturn);
    // v_los = R[0,2]*vx + R[1,2]*vy = -si * vabs * cos(theta); cos = rx/r
    float vlos = -si * vabs * rx * rinv;
    float zh = rz * ihz;
    // log2 of intensity, folded into the Gaussian exponent
    float ei = fmaf(-0.5f * zh, zh, -r * irdisk) * LN2_INV;
#pragma unroll
    for (int j = 0; j < 32; ++j) {
      float d = vlos - vel[j];
      acc[j] += fast_exp2(fmaf(kexp, d * d, ei));
    }
  }

  float norm = rsqrtf(TWO_PI * sig_sq);
  float scale = 0.25f * norm;                  // fold freq-mean (/4) + norm
  float* row = mid + x * NHI + y;              // mid layout: [F][x][y]
#pragma unroll
  for (int k = 0; k < 8; ++k) {
    float v = scale * (acc[4*k] + acc[4*k+1] + acc[4*k+2] + acc[4*k+3]);
    row[(c * 8 + k) * (NHI * NHI)] = v;
  }
}

// ---------------------------------------------------------------------------
// WMMA f32 16x16x4 operand synthesis per ISA 7.12.2 (wave32):
//   A 16x4 : VGPR0 = K{0|2}, VGPR1 = K{1|3}; lanes 0-15 M=row, 16-31 same rows
//   B 4x16 : mirrored, lanes carry N
//   C/D    : VGPR r -> rows r (lanes 0-15) / r+8 (lanes 16-31), N = lane%16
// P[y, Yg] = 0.5 iff (y >> 1) == Yg  (2:1 averaging matrix)
// ---------------------------------------------------------------------------

// Stage 2: tmp[F][x][Y] = sum_y mid[F][x][y] * P[y,Y]   (contract over y)
__global__ __launch_bounds__(32)
void ds_pass_y(const float* __restrict__ mid, float* __restrict__ tmp)
{
  int t  = blockIdx.x;               // 32F * 8 xTiles * 4 YTiles = 1024
  int Ft = t >> 5;
  int xT = (t >> 2) & 7;
  int YT = t & 3;
  int x0 = xT * 16, Y0 = YT * 16;

  int lane = threadIdx.x;
  int lm = lane & 15, lh = lane >> 4;

  const float* Mf = mid + Ft * (NHI * NHI);
  v8f acc = {0.f, 0.f, 0.f, 0.f, 0.f, 0.f, 0.f, 0.f};

  for (int kk = 0; kk < 32; ++kk) {
    int yb = kk * 4 + 2 * lh;                       // this lane's K pair
    const float* ap = Mf + (x0 + lm) * NHI + yb;    // A = mid tile (8B aligned)
    v2f a; a.x = ap[0]; a.y = ap[1];
    // B = P tile; (yb>>1) == (yb+1)>>1 == 2kk+lh, so both halves equal
    float pv = ((2 * kk + lh) == (Y0 + lm)) ? 0.5f : 0.0f;
    v2f b; b.x = pv; b.y = pv;
    acc = __builtin_amdgcn_wmma_f32_16x16x4_f32(
        false, a, false, b, (short)0, acc, false, false);
  }

  float* Tf = tmp + Ft * (NHI * NOUT);
#pragma unroll
  for (int r = 0; r < 8; ++r)
    Tf[(x0 + r + 8 * lh) * NOUT + Y0 + lm] = acc[r];
}

// Stage 3: out[F][X][Y] = sum_x P[x,X] * tmp[F][x][Y]   (contract over x)
__global__ __launch_bounds__(32)
void ds_pass_x(const float* __restrict__ tmp, float* __restrict__ out)
{
  int t  = blockIdx.x;               // 32F * 4 XTiles * 4 YTiles = 512
  int Ft = t >> 4;
  int XT = (t >> 2) & 3;
  int YT = t & 3;
  int X0 = XT * 16, Y0 = YT * 16;

  int lane = threadIdx.x;
  int lm = lane & 15, lh = lane >> 4;

  const float* Tf = tmp + Ft * (NHI * NOUT);
  v8f acc = {0.f, 0.f, 0.f, 0.f, 0.f, 0.f, 0.f, 0.f};

  for (int kk = 0; kk < 32; ++kk) {
    int xb = kk * 4 + 2 * lh;
    // A = P^T tile (in registers): P[4kk+k, X0+m], (4kk+k)>>1 == 2kk+lh
    float pv = ((2 * kk + lh) == (X0 + lm)) ? 0.5f : 0.0f;
    v2f a; a.x = pv; a.y = pv;
    // B = tmp tile: rows xb, xb+1 at column Y0+lm
    v2f b;
    b.x = Tf[xb * NOUT + Y0 + lm];
    b.y = Tf[(xb + 1) * NOUT + Y0 + lm];
    acc = __builtin_amdgcn_wmma_f32_16x16x4_f32(
        false, a, false, b, (short)0, acc, false, false);
  }

  float* Of = out + Ft * (NOUT * NOUT);
#pragma unroll
  for (int r = 0; r < 8; ++r)
    Of[(X0 + r + 8 * lh) * NOUT + Y0 + lm] = acc[r];
}

// ---------------------------------------------------------------------------
extern "C" void kernel_launch(void* const* d_in, const int* in_sizes, int n_in,
                              void* d_out, int out_size, void* d_ws, size_t ws_size,
                              hipStream_t stream)
{
  const float* inc   = (const float*)d_in[0];
  const float* rot   = (const float*)d_in[1];
  const float* sig   = (const float*)d_in[2];
  const float* vsh   = (const float*)d_in[3];
  const float* vmax  = (const float*)d_in[4];
  const float* rturn = (const float*)d_in[5];
  const float* rdisk = (const float*)d_in[6];
  const float* hz    = (const float*)d_in[7];
  const float* freqs = (const float*)d_in[8];
  float* out = (float*)d_out;

  float* mid = (float*)d_ws;                         // 32*128*128 f32 = 2 MB
  float* tmp = mid + NFREQ * NHI * NHI;              // 32*128*64  f32 = 1 MB

  // 128*128 pixels * 4 channel-chunks = 65536 threads
  cube_stage1<<<256, 256, 0, stream>>>(inc, rot, sig, vsh, vmax, rturn,
                                       rdisk, hz, freqs, mid);
  ds_pass_y<<<1024, 32, 0, stream>>>(mid, tmp);
  ds_pass_x<<<512, 32, 0, stream>>>(tmp, out);
}